// DFBBlock_72121090834963
// MI455X (gfx1250) — compile-verified
//
#include <hip/hip_runtime.h>
#include <math.h>

typedef __attribute__((ext_vector_type(2))) float v2f;
typedef __attribute__((ext_vector_type(8))) float v8f;

#define IMG_H 256
#define IMG_W 256
#define HW (IMG_H * IMG_W)
#define NF 64
#define NB 8

__device__ __forceinline__ float xat(const float* __restrict__ X, int yy, int xx) {
  return (yy >= 0 && yy < IMG_H && xx >= 0 && xx < IMG_W) ? X[(yy << 8) + xx] : 0.f;
}

// Ring index for out-of-grid intermediate positions (coords in {-1,256} on one axis)
__device__ __forceinline__ int ring_r(int qy, int qx) {
  if (qy == -1)    return qx + 1;        // qx in [-1,256] -> [0,257]
  if (qy == IMG_H) return qx + 259;      // -> [258,515]
  if (qx == -1)    return 516 + qy;      // qy in [0,255] -> [516,771]
  return 772 + qy;                       // -> [772,1027]
}

// ---------------------------------------------------------------------------
// Kernel 1: power iteration for ||A^T A|| on a 256x256 image, single WGP,
// image resident in LDS.  Writes {lip2, sigma, tau} to scal[0..2].
// ---------------------------------------------------------------------------
__global__ __launch_bounds__(1024) void power_iter_kernel(
    const float* __restrict__ w, const float* __restrict__ beta,
    const float* __restrict__ log_sigma, float* __restrict__ scal) {
  extern __shared__ float smem[];
  float* X   = smem;              // 65536 : image
  float* SV  = X + HW;            // 1028*9: ring boundary products
  float* Wm  = SV + 1028 * 9;     // 576   : weights
  float* Gm  = Wm + 576;          // 25    : effective 5x5 kernel
  float* RED = Gm + 25;           // 33    : reduction scratch

  const int tid = threadIdx.x;
  for (int i = tid; i < 576; i += 1024) Wm[i] = w[i];
#pragma unroll
  for (int k = 0; k < 64; ++k) X[tid + (k << 10)] = 1.0f / 256.0f;  // ones/||ones||
  __syncthreads();
  if (tid < 25) {
    int dy = tid / 5 - 2, dx = tid % 5 - 2;
    float g = 0.f;
    for (int f = 0; f < NF; ++f)
      for (int i = 0; i < 3; ++i)
        for (int j = 0; j < 3; ++j) {
          int i2 = i + dy, j2 = j + dx;
          if (i2 >= 0 && i2 < 3 && j2 >= 0 && j2 < 3)
            g += Wm[f * 9 + i * 3 + j] * Wm[f * 9 + i2 * 3 + j2];
        }
    Gm[tid] = g;
  }
  __syncthreads();

  const int px  = tid & 255;  // fixed column per thread
  const int py0 = tid >> 8;   // pixel p = tid + k*1024 -> row py0 + 4k
  float val = 1.0f;
  float yv[64];

  for (int iter = 0; iter < 300; ++iter) {
    // --- Phase A: products for virtual (out-of-grid) intermediate positions
    for (int r = tid; r < 1028; r += 1024) {
      int qy, qx;
      if (r < 258)      { qy = -1;    qx = r - 1;   }
      else if (r < 516) { qy = IMG_H; qx = r - 259; }
      else if (r < 772) { qx = -1;    qy = r - 516; }
      else              { qx = IMG_W; qy = r - 772; }
      float s[9];
#pragma unroll
      for (int t = 0; t < 9; ++t) s[t] = 0.f;
      for (int f = 0; f < NF; ++f) {
        float cv = 0.f;
#pragma unroll
        for (int a = 0; a < 3; ++a)
#pragma unroll
          for (int b2 = 0; b2 < 3; ++b2)
            cv += Wm[f * 9 + a * 3 + b2] * xat(X, qy + a - 1, qx + b2 - 1);
#pragma unroll
        for (int t = 0; t < 9; ++t) s[t] += Wm[f * 9 + t] * cv;
      }
#pragma unroll
      for (int t = 0; t < 9; ++t) SV[r * 9 + t] = s[t];
    }
    __syncthreads();

    // --- Phase B: y = G*x (5x5) minus exact boundary correction
    float sumsq = 0.f;
#pragma unroll
    for (int k = 0; k < 64; ++k) {
      int py = py0 + (k << 2);
      float acc = 0.f;
#pragma unroll
      for (int dy = -2; dy <= 2; ++dy)
#pragma unroll
        for (int dx = -2; dx <= 2; ++dx)
          acc += Gm[(dy + 2) * 5 + (dx + 2)] * xat(X, py + dy, px + dx);
      if (py == 0 || py == IMG_H - 1 || px == 0 || px == IMG_W - 1) {
#pragma unroll
        for (int dy = -1; dy <= 1; ++dy)
#pragma unroll
          for (int dx = -1; dx <= 1; ++dx) {
            int qy = py + dy, qx = px + dx;
            if (qy < 0 || qy >= IMG_H || qx < 0 || qx >= IMG_W)
              acc -= SV[ring_r(qy, qx) * 9 + (1 - dy) * 3 + (1 - dx)];
          }
      }
      yv[k] = acc;
      sumsq += acc * acc;
    }
    // --- deterministic norm reduction (wave32 shuffle + serial over waves)
#pragma unroll
    for (int off = 16; off > 0; off >>= 1) sumsq += __shfl_down(sumsq, off, 32);
    if ((tid & 31) == 0) RED[tid >> 5] = sumsq;
    __syncthreads();
    if (tid == 0) {
      float tot = 0.f;
      for (int i = 0; i < 32; ++i) tot += RED[i];
      RED[32] = sqrtf(tot);
    }
    __syncthreads();
    float v2  = RED[32];
    float rel = fabsf(v2 - val) / val;
    val = v2;
    float inv = 1.0f / v2;
#pragma unroll
    for (int k = 0; k < 64; ++k) X[tid + (k << 10)] = yv[k] * inv;
    if (rel < 1e-4f) break;   // uniform across all threads
    __syncthreads();
  }

  if (tid == 0) {
    float lip2 = val;
    float sg   = expf(log_sigma[0]) + 0.05f;
    scal[0] = lip2;
    scal[1] = sg;
    scal[2] = 0.99f / (beta[0] * 0.5f + sg * lip2);
  }
}

// ---------------------------------------------------------------------------
// Kernel 2: output0 = clip(x - tau*x*HtH - tau*conv_t(u_in,w) + tau*bias, 0, 1)
// One thread per pixel; bandwidth bound on u_in (134 MB).
// ---------------------------------------------------------------------------
__global__ __launch_bounds__(256) void dx_kernel(
    const float* __restrict__ x_in, const float* __restrict__ u_in,
    const float* __restrict__ bias, const float* __restrict__ HtH,
    const float* __restrict__ w, const float* __restrict__ scal,
    float* __restrict__ out0) {
  __shared__ float Wm[576];
  const int tid = threadIdx.x;
  for (int i = tid; i < 576; i += 256) Wm[i] = w[i];
  __syncthreads();

  const int y = blockIdx.x & 255;
  const int b = blockIdx.x >> 8;
  const int x = tid;
  const float tau = scal[2];

  float ct = 0.f;
  const float* ub = u_in + (size_t)b * NF * HW;
  for (int f = 0; f < NF; ++f) {
    const float* uf = ub + (size_t)f * HW;
#pragma unroll
    for (int i = 0; i < 3; ++i) {
      int yy = y + 1 - i;                 // conv_t: sum_f sum_ij w[f,i,j]*u[f,y+1-i,x+1-j]
      if (yy < 0 || yy >= IMG_H) continue;
#pragma unroll
      for (int j = 0; j < 3; ++j) {
        int xx = x + 1 - j;
        if (xx < 0 || xx >= IMG_W) continue;
        ct += Wm[f * 9 + i * 3 + j] * uf[(yy << 8) + xx];
      }
    }
  }
  int p = (y << 8) + x;
  size_t gp = (size_t)b * HW + p;
  float xv = x_in[gp];
  float Dx = xv - tau * xv * HtH[p] - tau * ct + tau * bias[gp];
  out0[gp] = fminf(fmaxf(Dx, 0.f), 1.f);
}

// ---------------------------------------------------------------------------
// Kernel 3: output1 = clip(conv(2*sigma*out0 - sigma*x_in, w) + u_in, -l, l)
// Implicit GEMM on V_WMMA_F32_16X16X4_F32: M=16 px, N=16 ch (x4), K=9 pad 12.
// ---------------------------------------------------------------------------
__global__ __launch_bounds__(256) void du_wmma_kernel(
    const float* __restrict__ x_in, const float* __restrict__ u_in,
    const float* __restrict__ out0, const float* __restrict__ w,
    const float* __restrict__ lambd, const float* __restrict__ scal,
    float* __restrict__ out1) {
  __shared__ float tT[3][132];        // fused-input halo tile (rows y-1..y+1)
  __shared__ float wB[12 * 64];       // weights as B matrix [K=12][N=64], K>=9 zero
  __shared__ float stage[NF * 128];   // D staging for coalesced writeback

  const int tid  = threadIdx.x;
  const int blk  = blockIdx.x;
  const int xblk = blk & 1;
  const int y    = (blk >> 1) & 255;
  const int b    = blk >> 9;
  const int X0   = xblk << 7;
  const float sigma = scal[1];

  for (int idx = tid; idx < 3 * 130; idx += 256) {
    int r = idx / 130, c = idx - r * 130;
    int yy = y - 1 + r, xx = X0 - 1 + c;
    float v = 0.f;
    if (yy >= 0 && yy < IMG_H && xx >= 0 && xx < IMG_W) {
      size_t off = (size_t)b * HW + (yy << 8) + xx;
      v = sigma * (2.f * out0[off] - x_in[off]);   // conv linearity fuses both convs
    }
    tT[r][c] = v;
  }
  for (int idx = tid; idx < 12 * 64; idx += 256) {
    int k = idx >> 6, n = idx & 63;
    wB[idx] = (k < 9) ? w[n * 9 + k] : 0.f;
  }
  __syncthreads();

  const int wv   = tid >> 5;        // wave -> 16-pixel M tile
  const int lane = tid & 31;
  const int nl   = lane & 15;
  const int khi  = lane >> 4;       // K-half select per f32 A/B layout
  const int xb   = (wv << 4) + nl;  // pixel (M) position within 128-wide tile

  // A fragments: A[m][k] = t(y + k/3 - 1, X0 + m + k%3 - 1); lane holds K {2khi, 2khi+1}
  v2f a[3];
#pragma unroll
  for (int s = 0; s < 3; ++s) {
    int k0 = 4 * s + 2 * khi;
    int k1 = k0 + 1;
    float a0 = 0.f, a1 = 0.f;
    if (k0 < 9) a0 = tT[k0 / 3][xb + k0 % 3];
    if (k1 < 9) a1 = tT[k1 / 3][xb + k1 % 3];
    a[s].x = a0;
    a[s].y = a1;
  }

#pragma unroll
  for (int nt = 0; nt < 4; ++nt) {  // 4 x 16-channel N tiles
    v8f acc = {};
#pragma unroll
    for (int s = 0; s < 3; ++s) {
      int row = (4 * s + 2 * khi) * 64 + (nt << 4) + nl;
      v2f bf;
      bf.x = wB[row];
      bf.y = wB[row + 64];
      acc = __builtin_amdgcn_wmma_f32_16x16x4_f32(
          false, a[s], false, bf, (short)0, acc, false, false);
    }
    // D layout: VGPR j -> M = j + 8*khi, N = nl
#pragma unroll
    for (int j = 0; j < 8; ++j)
      stage[((nt << 4) + nl) * 128 + (wv << 4) + j + 8 * khi] = acc[j];
  }
  __syncthreads();

  const float lam = lambd[0];
  const size_t base = (size_t)b * NF * HW + ((size_t)y << 8) + X0;
  for (int idx = tid; idx < NF * 128; idx += 256) {
    int ch = idx >> 7, p = idx & 127;
    size_t g = base + (size_t)ch * HW + p;
    float v = stage[idx] + u_in[g];
    v = fminf(fmaxf(v, -lam), lam);
    out1[g] = v;
  }
}

// ---------------------------------------------------------------------------
extern "C" void kernel_launch(void* const* d_in, const int* in_sizes, int n_in,
                              void* d_out, int out_size, void* d_ws, size_t ws_size,
                              hipStream_t stream) {
  const float* x_in  = (const float*)d_in[0];
  const float* u_in  = (const float*)d_in[1];
  const float* bias  = (const float*)d_in[2];
  const float* beta  = (const float*)d_in[3];
  const float* lambd = (const float*)d_in[4];
  const float* HtH   = (const float*)d_in[5];
  const float* w     = (const float*)d_in[6];
  const float* lsig  = (const float*)d_in[7];

  float* out0 = (float*)d_out;           // [8,1,256,256]
  float* out1 = out0 + (size_t)NB * HW;  // [8,64,256,256]
  float* scal = (float*)d_ws;            // {lip2, sigma, tau}

  size_t smem = (size_t)(HW + 1028 * 9 + 576 + 25 + 33) * sizeof(float); // ~295 KB
  power_iter_kernel<<<1, 1024, smem, stream>>>(w, beta, lsig, scal);
  dx_kernel<<<NB * IMG_H, 256, 0, stream>>>(x_in, u_in, bias, HtH, w, scal, out0);
  du_wmma_kernel<<<NB * IMG_H * 2, 256, 0, stream>>>(x_in, u_in, out0, w, lambd, scal, out1);
}